// KANLayer_63479616635239
// MI455X (gfx1250) — compile-verified
//
#include <hip/hip_runtime.h>
#include <hip/hip_bf16.h>

typedef __attribute__((ext_vector_type(2))) float v2f;
typedef __attribute__((ext_vector_type(8))) float v8f;

// Problem constants (match reference)
#define BSZ 1024
#define IDIM 256
#define ODIM 256
#define KNOTS 128

#define BT 16        // batch tile per block
#define OT 64        // output tile per block (4 waves x 16)
#define NWAVES 4
#define NTHREADS (NWAVES * 32)
#define LSTR (IDIM + 4)   // padded LDS stride (bank-conflict free for A-frag loads)

// ---------------------------------------------------------------------------
// Kernel 1: per-element activation / knot preprocessing
//   sact = silu(tanh(x));  scaled = (clip(tanh(x),-1,1)+1)*63.5
//   l2 = clamp(floor(scaled), 0, 126);  frac = scaled - l2   (in [0,1])
// The l2<=126 clamp + frac reaching 1.0 reproduces right=min(left+1,127)
// (left==127 only occurs at scaled==127.0 exactly, where frac==0).
// ---------------------------------------------------------------------------
__global__ __launch_bounds__(256) void kan_act(
    const float* __restrict__ x,
    float* __restrict__ sact,
    float* __restrict__ frac,
    unsigned char* __restrict__ l2idx,
    int n)
{
    int idx = blockIdx.x * blockDim.x + threadIdx.x;
    if (idx >= n) return;
    float p = tanhf(x[idx]);
    float s = p / (1.0f + __expf(-p));          // silu(p)
    float cc = fminf(fmaxf(p, -1.0f), 1.0f);
    float scaled = (cc + 1.0f) * 63.5f;          // (c - DMIN)/step, step = 2/127
    int l = (int)floorf(scaled);
    l = l < 0 ? 0 : (l > 126 ? 126 : l);
    float f = scaled - (float)l;
    sact[idx]  = s;
    frac[idx]  = f;
    l2idx[idx] = (unsigned char)l;
}

// ---------------------------------------------------------------------------
// Kernel 2: fused base-GEMM (f32 WMMA) + spline gather/lerp + bias
// Block: 128 threads = 4 waves. Each wave computes one 16x16 (b x o) tile.
// All waves share the b-tile activation data staged in LDS.
// ---------------------------------------------------------------------------
__global__ __launch_bounds__(NTHREADS) void kan_main(
    const float* __restrict__ W,       // (O, I)
    const float* __restrict__ coeff,   // (O, I, K)
    const float* __restrict__ scal,    // (O, I)
    const float* __restrict__ bias,    // (O)
    const float* __restrict__ sact,    // (B, I)
    const float* __restrict__ frac,    // (B, I)
    const unsigned char* __restrict__ l2idx, // (B, I)
    float* __restrict__ out)           // (B, O)
{
    __shared__ float         s_act[BT * LSTR];
    __shared__ float         s_frac[BT * LSTR];
    __shared__ unsigned char s_l2[BT * LSTR];

    const int tid  = threadIdx.x;
    const int b0   = blockIdx.x * BT;

    // ---- stage b-tile activation data into LDS (uniform, no divergence) ----
    for (int idx = tid; idx < BT * IDIM; idx += NTHREADS) {
        int bb = idx / IDIM;
        int ii = idx - bb * IDIM;
        int g  = (b0 + bb) * IDIM + ii;
        s_act [bb * LSTR + ii] = sact[g];
        s_frac[bb * LSTR + ii] = frac[g];
        s_l2  [bb * LSTR + ii] = l2idx[g];
    }
    __syncthreads();

    const int wave = tid >> 5;
    const int lane = tid & 31;
    const int n    = lane & 15;   // N index within 16x16 tile
    const int hi   = lane >> 4;   // 0: lanes 0-15, 1: lanes 16-31
    const int o0   = blockIdx.y * OT + wave * 16;
    const int o    = o0 + n;

    const float*         Wrow  = W     + (size_t)o * IDIM;
    const float*         crow  = coeff + (size_t)o * (IDIM * KNOTS);
    const float*         srow  = scal  + (size_t)o * IDIM;
    const float*         arow  = s_act  + n * LSTR;   // A-frag row (M = lane&15)

    v8f  acc = {};                 // WMMA base accumulator
    float sp[8];
    #pragma unroll
    for (int r = 0; r < 8; ++r) sp[r] = 0.0f;

    const int koff = hi * 2;       // K sub-offset for A/B fragments

    for (int i0 = 0; i0 < IDIM; i0 += 4) {
        // ---- base GEMM step: D += A(16x4) x B(4x16), f32 WMMA ----
        v2f a, b;
        a.x = arow[i0 + koff];
        a.y = arow[i0 + koff + 1];
        b.x = Wrow[i0 + koff];
        b.y = Wrow[i0 + koff + 1];
        acc = __builtin_amdgcn_wmma_f32_16x16x4_f32(
                  false, a, false, b, (short)0, acc, false, false);

        // ---- spline: 2-point gather + lerp for i0..i0+3 ----
        #pragma unroll
        for (int ii = i0; ii < i0 + 4; ++ii) {
            const float sc = srow[ii];
            const float* cbase = crow + ii * KNOTS;
            #pragma unroll
            for (int r = 0; r < 8; ++r) {
                const int m   = r + hi * 8;          // matches C/D VGPR layout
                const int li  = m * LSTR + ii;       // broadcast LDS reads
                const int l   = (int)s_l2[li];
                const float f = s_frac[li];
                const float cl = cbase[l];
                const float cr = cbase[l + 1];
                sp[r] += (cl + f * (cr - cl)) * sc;
            }
        }
    }

    // ---- epilogue: base + spline + bias ----
    const float bo = bias[o];
    #pragma unroll
    for (int r = 0; r < 8; ++r) {
        const int m = r + hi * 8;
        out[(size_t)(b0 + m) * ODIM + o] = acc[r] + sp[r] * 1.0f + bo;
    }
}

// ---------------------------------------------------------------------------
extern "C" void kernel_launch(void* const* d_in, const int* in_sizes, int n_in,
                              void* d_out, int out_size, void* d_ws, size_t ws_size,
                              hipStream_t stream) {
    const float* x     = (const float*)d_in[0];   // (B, I)
    const float* W     = (const float*)d_in[1];   // (O, I)
    const float* coeff = (const float*)d_in[2];   // (O, I, K)
    const float* scal  = (const float*)d_in[3];   // (O, I)
    const float* bias  = (const float*)d_in[4];   // (O)
    float* out = (float*)d_out;

    const int BI = BSZ * IDIM;
    float* w_sact = (float*)d_ws;
    float* w_frac = w_sact + BI;
    unsigned char* w_l2 = (unsigned char*)(w_frac + BI);

    kan_act<<<(BI + 255) / 256, 256, 0, stream>>>(x, w_sact, w_frac, w_l2, BI);

    dim3 grid(BSZ / BT, ODIM / OT);
    kan_main<<<grid, NTHREADS, 0, stream>>>(W, coeff, scal, bias,
                                            w_sact, w_frac, w_l2, out);
}